// GNNModel_8830452761311
// MI455X (gfx1250) — compile-verified
//
#include <hip/hip_runtime.h>
#include <hip/hip_bf16.h>

typedef float v2f __attribute__((ext_vector_type(2)));
typedef float v8f __attribute__((ext_vector_type(8)));

// ---------------------------------------------------------------------------
// Utility fills
// ---------------------------------------------------------------------------
__global__ void fill_f32(float* __restrict__ p, float v, long long n) {
    long long t = (long long)blockIdx.x * blockDim.x + threadIdx.x;
    if (t < n) p[t] = v;
}

// deg[dst[e]] += 1  (deg pre-initialized to 1.0 for the self-loop)
__global__ void deg_kernel(const int* __restrict__ dst, float* __restrict__ deg, int E) {
    int e = blockIdx.x * blockDim.x + threadIdx.x;
    if (e < E) atomicAdd(&deg[dst[e]], 1.0f);
}

// deg -> dinv in place
__global__ void rsqrt_kernel(float* __restrict__ deg, int n) {
    int i = blockIdx.x * blockDim.x + threadIdx.x;
    if (i < n) {
        float d = deg[i];
        deg[i] = (d > 0.0f) ? rsqrtf(d) : 0.0f;
    }
}

// norm[e] = dinv[src[e]] * dinv[dst[e]]
__global__ void norm_kernel(const int* __restrict__ src, const int* __restrict__ dst,
                            const float* __restrict__ dinv, float* __restrict__ norm, int E) {
    int e = blockIdx.x * blockDim.x + threadIdx.x;
    if (e < E) norm[e] = dinv[src[e]] * dinv[dst[e]];
}

// ---------------------------------------------------------------------------
// fp32 WMMA GEMM:  Y[nrows, NOUT] = X[nrows, K] @ W[K, NOUT]
// One wave (32 lanes) computes one 16x16 output tile using
// V_WMMA_F32_16X16X4_F32 over K/4 steps.  nrows must be a multiple of 16.
//
// Fragment layouts per CDNA5 ISA 7.12.2:
//   A (16x4 f32): lanes 0-15 = rows, half = lane>>4 selects K pair;
//                 v0 = K(2*half), v1 = K(2*half+1)
//   B (4x16 f32): lanes 0-15 = cols (mirrored layout)
//   C/D (16x16 f32, 8 VGPRs): vgpr r holds M = r + 8*half, N = lane&15
// ---------------------------------------------------------------------------
template <int K, int NOUT>
__global__ void wmma_gemm_f32(const float* __restrict__ X, const float* __restrict__ W,
                              float* __restrict__ Y, int nrows) {
    const int wavesPerBlock = blockDim.x >> 5;
    const int waveId = blockIdx.x * wavesPerBlock + (threadIdx.x >> 5);
    const int nTiles = NOUT / 16;
    const int mTile = waveId / nTiles;
    const int nTile = waveId % nTiles;
    if (mTile * 16 >= nrows) return;     // wave-uniform: EXEC stays all-ones

    const int lane = threadIdx.x & 31;
    const int half = lane >> 4;          // 0 or 1
    const int l16  = lane & 15;
    const int arow = mTile * 16 + l16;   // A row for this lane
    const int ncol = nTile * 16 + l16;   // output column for this lane

    v8f c = {0.f, 0.f, 0.f, 0.f, 0.f, 0.f, 0.f, 0.f};

    const float* __restrict__ xrow = X + (long long)arow * K;
#pragma unroll 4
    for (int k0 = 0; k0 < K; k0 += 4) {
        const int ka = k0 + half * 2;
        v2f a, b;
        a.x = xrow[ka];
        a.y = xrow[ka + 1];
        b.x = W[(long long)ka * NOUT + ncol];
        b.y = W[(long long)(ka + 1) * NOUT + ncol];
        c = __builtin_amdgcn_wmma_f32_16x16x4_f32(
                /*neg_a=*/false, a, /*neg_b=*/false, b,
                /*c_mod=*/(short)0, c, /*reuse_a=*/false, /*reuse_b=*/false);
    }

#pragma unroll
    for (int r = 0; r < 8; ++r) {
        const int row = mTile * 16 + half * 8 + r;
        Y[(long long)row * NOUT + ncol] = c[r];
    }
}

// ---------------------------------------------------------------------------
// Edge scatter: acc[dst[e], f] += H[src[e], f] * norm[e]
// thread t -> (edge = t/F, f = t%F): each wave touches 32 contiguous floats
// of one edge's row (coalesced 128B), rows land in the 192MB L2.
// ---------------------------------------------------------------------------
template <int F>
__global__ void scatter_edges(const float* __restrict__ H, const int* __restrict__ src,
                              const int* __restrict__ dst, const float* __restrict__ norm,
                              float* __restrict__ acc, int E) {
    long long t = (long long)blockIdx.x * blockDim.x + threadIdx.x;
    long long total = (long long)E * F;
    if (t >= total) return;
    const int e = (int)(t / F);
    const int f = (int)(t % F);
    const int s = src[e];
    const int d = dst[e];
    const float v = H[(long long)s * F + f] * norm[e];
    atomicAdd(&acc[(long long)d * F + f], v);
}

// acc[i,f] = act(acc[i,f] + H[i,f]*dinv[i]^2 + bias[f])   (self-loop + bias)
template <int F, bool RELU>
__global__ void finalize_kernel(float* __restrict__ acc, const float* __restrict__ H,
                                const float* __restrict__ dinv, const float* __restrict__ bias,
                                int n) {
    long long t = (long long)blockIdx.x * blockDim.x + threadIdx.x;
    long long total = (long long)n * F;
    if (t >= total) return;
    const int i = (int)(t / F);
    const int f = (int)(t % F);
    const float di = dinv[i];
    float v = acc[t] + H[t] * (di * di) + bias[f];
    if (RELU) v = (v > 0.0f) ? v : 0.0f;
    acc[t] = v;
}

// ---------------------------------------------------------------------------
// Launch
// ---------------------------------------------------------------------------
extern "C" void kernel_launch(void* const* d_in, const int* in_sizes, int n_in,
                              void* d_out, int out_size, void* d_ws, size_t ws_size,
                              hipStream_t stream) {
    constexpr int IN_DIM = 128, HID_DIM = 64, OUT_DIM = 32;

    const float* x  = (const float*)d_in[0];
    const int*   ei = (const int*)d_in[1];     // [2, E] flattened
    const float* W1 = (const float*)d_in[2];
    const float* b1 = (const float*)d_in[3];
    const float* W2 = (const float*)d_in[4];
    const float* b2 = (const float*)d_in[5];

    const int N = in_sizes[0] / IN_DIM;        // 100000
    const int E = in_sizes[1] / 2;             // 1600000
    const int* src = ei;
    const int* dst = ei + E;
    float* out = (float*)d_out;

    // Workspace carve-up (256B aligned)
    char* ws = (char*)d_ws;
    size_t off = 0;
    auto alignup = [](size_t v) { return (v + 255) & ~(size_t)255; };
    float* dinv = (float*)(ws + off); off += alignup((size_t)N * 4);
    float* norm = (float*)(ws + off); off += alignup((size_t)E * 4);
    float* h1   = (float*)(ws + off); off += alignup((size_t)N * HID_DIM * 4);
    float* acc1 = (float*)(ws + off); off += alignup((size_t)N * HID_DIM * 4);
    float* g2   = (float*)(ws + off); off += alignup((size_t)N * OUT_DIM * 4);
    (void)ws_size; (void)n_in; (void)out_size;

    const int B = 256;
    auto blocks = [&](long long n) { return (unsigned)((n + B - 1) / B); };

    // 1) init: deg=1 (self-loop), acc1=0, out=0
    fill_f32<<<blocks(N), B, 0, stream>>>(dinv, 1.0f, N);
    fill_f32<<<blocks((long long)N * HID_DIM), B, 0, stream>>>(acc1, 0.0f, (long long)N * HID_DIM);
    fill_f32<<<blocks((long long)N * OUT_DIM), B, 0, stream>>>(out, 0.0f, (long long)N * OUT_DIM);

    // 2) degrees -> dinv -> per-edge norms (shared by both layers)
    deg_kernel<<<blocks(E), B, 0, stream>>>(dst, dinv, E);
    rsqrt_kernel<<<blocks(N), B, 0, stream>>>(dinv, N);
    norm_kernel<<<blocks(E), B, 0, stream>>>(src, dst, dinv, norm, E);

    // 3) layer 1: h1 = x @ W1 via fp32 WMMA
    {
        const int tiles = (N / 16) * (HID_DIM / 16);     // 6250 * 4
        const int wavesPerBlock = B / 32;                // 8
        wmma_gemm_f32<IN_DIM, HID_DIM>
            <<<(tiles + wavesPerBlock - 1) / wavesPerBlock, B, 0, stream>>>(x, W1, h1, N);
    }
    scatter_edges<HID_DIM><<<blocks((long long)E * HID_DIM), B, 0, stream>>>(h1, src, dst, norm, acc1, E);
    finalize_kernel<HID_DIM, true><<<blocks((long long)N * HID_DIM), B, 0, stream>>>(acc1, h1, dinv, b1, N);

    // 4) layer 2: g2 = relu(h) @ W2 via fp32 WMMA
    {
        const int tiles = (N / 16) * (OUT_DIM / 16);     // 6250 * 2
        const int wavesPerBlock = B / 32;
        wmma_gemm_f32<HID_DIM, OUT_DIM>
            <<<(tiles + wavesPerBlock - 1) / wavesPerBlock, B, 0, stream>>>(acc1, W2, g2, N);
    }
    scatter_edges<OUT_DIM><<<blocks((long long)E * OUT_DIM), B, 0, stream>>>(g2, src, dst, norm, out, E);
    finalize_kernel<OUT_DIM, false><<<blocks((long long)N * OUT_DIM), B, 0, stream>>>(out, g2, dinv, b2, N);
}